// GATTreeEncoder_63264868270303
// MI455X (gfx1250) — compile-verified
//
#include <hip/hip_runtime.h>
#include <cstdint>

// GAT tree encoder for MI455X (gfx1250, wave32).
// GEMMs: V_WMMA_F32_16X16X4_F32 (full fp32) with the B panel staged into LDS
// via GLOBAL_LOAD_ASYNC_TO_LDS_B32 (ASYNCcnt path) when available.
// Edge softmax/aggregation: L2-resident scatter/gather with f32 atomics.

typedef __attribute__((ext_vector_type(2))) float v2f;
typedef __attribute__((ext_vector_type(8))) float v8f;

#if defined(__has_builtin)
#if __has_builtin(__builtin_amdgcn_global_load_async_to_lds_b32)
#define HAVE_ASYNC_LDS 1
#endif
#endif

typedef __attribute__((address_space(1))) int* as1_ip;   // global (AS1) int*
typedef __attribute__((address_space(3))) int* as3_ip;   // LDS (AS3) int*

__device__ __forceinline__ void wait_asynccnt0() {
#if defined(__has_builtin) && __has_builtin(__builtin_amdgcn_s_wait_asynccnt)
    __builtin_amdgcn_s_wait_asynccnt(0);
#else
    asm volatile("s_wait_asynccnt 0x0" ::: "memory");
#endif
}

// ---------------------------------------------------------------------------
// WMMA GEMM: out = act(A[M,K] @ B[K,N] + bias). One wave per 16x16 tile.
// The block's K x 16 B panel is staged once into LDS (async) and shared by
// all 8 waves. M multiple of 16 (50000, 64 here), K multiple of 4, K <= 128.
// ---------------------------------------------------------------------------
__global__ void gemm_wmma_f32(const float* __restrict__ A,
                              const float* __restrict__ B,
                              const float* __restrict__ bias,
                              float* __restrict__ out,
                              int M, int K, int Nout, int relu)
{
    __shared__ float Bs[128 * 16];            // K<=128 rows x 16 cols, 8KB

    const int tid   = threadIdx.x;
    const int wave  = tid >> 5;
    const int lane  = tid & 31;
    const int tileN = blockIdx.y;
    const int col0  = tileN << 4;

    // Cooperative stage of B[:, col0:col0+16) into LDS. All waves participate
    // (before any early return) so the barrier is safe.
    for (int idx = tid; idx < K * 16; idx += 256) {
        const int k = idx >> 4, n = idx & 15;
#ifdef HAVE_ASYNC_LDS
        __builtin_amdgcn_global_load_async_to_lds_b32(
            (as1_ip)(B + (size_t)k * Nout + col0 + n),
            (as3_ip)(Bs + idx), 0, 0);
#else
        Bs[idx] = B[(size_t)k * Nout + col0 + n];
#endif
    }
#ifdef HAVE_ASYNC_LDS
    wait_asynccnt0();
#endif
    __syncthreads();

    const int mtiles = M >> 4;
    const int tileM  = blockIdx.x * 8 + wave;
    if (tileM >= mtiles) return;              // wave-uniform: EXEC stays all-1s

    const int half = lane >> 4;               // 0: lanes 0-15, 1: lanes 16-31
    const int l16  = lane & 15;
    const int row  = (tileM << 4) + l16;      // A row held by this lane
    const int col  = col0 + l16;              // B col held by this lane

    v8f c = {};
    for (int k0 = 0; k0 < K; k0 += 4) {
        const int kb = k0 + (half << 1);      // lanes16-31 hold K = k0+2, k0+3
        v2f a, b;
        a.x = A[(size_t)row * K + kb];
        a.y = A[(size_t)row * K + kb + 1];
        b.x = Bs[kb * 16 + l16];
        b.y = Bs[(kb + 1) * 16 + l16];
        c = __builtin_amdgcn_wmma_f32_16x16x4_f32(false, a, false, b,
                                                  (short)0, c, false, false);
    }
    // C/D layout: VGPR r -> M = r + 8*half, N = l16
    const float bv = bias ? bias[col] : 0.0f;
#pragma unroll
    for (int r = 0; r < 8; ++r) {
        const int orow = (tileM << 4) + r + (half << 3);
        float v = c[r] + bv;
        if (relu) v = v > 0.0f ? v : 0.0f;
        out[(size_t)orow * Nout + col] = v;
    }
}

// ---------------------------------------------------------------------------
// Per-node attention logits: als/ald[n*heads+h] = sum_c hW[n,h,c]*a{src,dst}[h,c]
// ---------------------------------------------------------------------------
__global__ void attn_coef(const float* __restrict__ hW,
                          const float* __restrict__ asrc,
                          const float* __restrict__ adst,
                          float* __restrict__ als, float* __restrict__ ald,
                          int N, int heads, int oc, int hshift)
{
    const int idx = blockIdx.x * blockDim.x + threadIdx.x;
    if (idx >= N * heads) return;
    const int n  = idx >> hshift;
    const int hd = idx & (heads - 1);
    const float* hp = hW + (size_t)n * 128 + hd * oc;
    const float* as = asrc + hd * oc;
    const float* ad = adst + hd * oc;
    float s1 = 0.0f, s2 = 0.0f;
    for (int c = 0; c < oc; ++c) { const float v = hp[c]; s1 += v * as[c]; s2 += v * ad[c]; }
    als[idx] = s1;
    ald[idx] = s2;
}

// Monotonic float<->uint mapping so atomicMax(u32) performs a float max.
__device__ __forceinline__ unsigned f32_ord(float f) {
    const unsigned u = __float_as_uint(f);
    return (u & 0x80000000u) ? ~u : (u | 0x80000000u);
}
__device__ __forceinline__ float ord_f32(unsigned s) {
    const unsigned u = (s & 0x80000000u) ? (s & 0x7fffffffu) : ~s;
    return __uint_as_float(u);
}

// Pass A: e = leaky_relu(als[src]+ald[dst]); segment max over dst.
__global__ void edge_max(const int* __restrict__ ei, int E, int Ep,
                         const float* __restrict__ als, const float* __restrict__ ald,
                         unsigned* __restrict__ menc, float* __restrict__ ebuf,
                         int heads, int hshift)
{
    const int idx = blockIdx.x * blockDim.x + threadIdx.x;
    if (idx >= Ep * heads) return;
    const int e  = idx >> hshift;
    const int hd = idx & (heads - 1);
    int s, d;
    if (e < E) { s = ei[e]; d = ei[E + e]; } else { s = e - E; d = s; }
    float v = als[s * heads + hd] + ald[d * heads + hd];
    v = v >= 0.0f ? v : 0.2f * v;             // leaky relu, slope 0.2
    ebuf[idx] = v;
    atomicMax(&menc[d * heads + hd], f32_ord(v));
}

// Pass B: t = exp(e - m[dst]); segment sum over dst; stash t in ebuf.
__global__ void edge_exp_sum(const int* __restrict__ ei, int E, int Ep,
                             const unsigned* __restrict__ menc,
                             float* __restrict__ ebuf, float* __restrict__ ssum,
                             int heads, int hshift)
{
    const int idx = blockIdx.x * blockDim.x + threadIdx.x;
    if (idx >= Ep * heads) return;
    const int e  = idx >> hshift;
    const int hd = idx & (heads - 1);
    const int d  = (e < E) ? ei[E + e] : (e - E);
    const float m = ord_f32(menc[d * heads + hd]);
    const float t = __expf(ebuf[idx] - m);
    ebuf[idx] = t;
    atomicAdd(&ssum[d * heads + hd], t);
}

// Pass C: acc[dst] += h[src] * alpha. One wave per edge, float4 per lane
// (32 lanes x 4ch = 128 channels). Gathers hit the 192MB L2 (h is 25.6MB).
__global__ void edge_aggregate(const int* __restrict__ ei, int E, int Ep,
                               const float* __restrict__ hW,
                               const float* __restrict__ ebuf,
                               const float* __restrict__ ssum,
                               float* __restrict__ acc,
                               int heads, int ocshift)
{
    const int gw   = (int)((blockIdx.x * blockDim.x + threadIdx.x) >> 5);
    const int lane = threadIdx.x & 31;
    if (gw >= Ep) return;                     // wave-uniform
    const int e = gw;
    int s, d;
    if (e < E) { s = ei[e]; d = ei[E + e]; } else { s = e - E; d = s; }
    __builtin_prefetch(hW + (size_t)s * 128 + (lane << 2), 0, 0);  // global_prefetch_b8
    const int cb = lane << 2;                 // channel base (4 per lane)
    const int hd = cb >> ocshift;             // head for these channels
    const float alpha = ebuf[e * heads + hd] / (ssum[d * heads + hd] + 1e-16f);
    const float4 hv = reinterpret_cast<const float4*>(hW)[(size_t)s * 32 + lane];
    float* ap = acc + (size_t)d * 128 + cb;
    atomicAdd(ap + 0, hv.x * alpha);
    atomicAdd(ap + 1, hv.y * alpha);
    atomicAdd(ap + 2, hv.z * alpha);
    atomicAdd(ap + 3, hv.w * alpha);
}

__global__ void bias_act(const float* __restrict__ acc, const float* __restrict__ b,
                         float* __restrict__ out, int total, int relu)
{
    const int i = blockIdx.x * blockDim.x + threadIdx.x;
    if (i >= total) return;
    float v = acc[i] + b[i & 127];
    if (relu) v = v > 0.0f ? v : 0.0f;
    out[i] = v;
}

__global__ void zero_u32(unsigned* __restrict__ p, int n)
{
    const int i = blockIdx.x * blockDim.x + threadIdx.x;
    if (i < n) p[i] = 0u;
}

__global__ void pool_sum(const float* __restrict__ feat, const int* __restrict__ batch,
                         float* __restrict__ psum, float* __restrict__ cnt, int N)
{
    const int i = blockIdx.x * blockDim.x + threadIdx.x;
    if (i >= N * 128) return;
    const int n = i >> 7, c = i & 127;
    const int g = batch[n];
    atomicAdd(&psum[g * 128 + c], feat[i]);
    if (c == 0) atomicAdd(&cnt[g], 1.0f);
}

__global__ void pool_div(const float* __restrict__ psum, const float* __restrict__ cnt,
                         float* __restrict__ pooled)
{
    const int i = blockIdx.x * blockDim.x + threadIdx.x;
    if (i >= 64 * 128) return;
    const int g = i >> 7;
    float c = cnt[g];
    c = c > 1.0f ? c : 1.0f;
    pooled[i] = psum[i] / c;
}

// ---------------------------------------------------------------------------
static inline unsigned cdiv(long long a, long long b) { return (unsigned)((a + b - 1) / b); }

extern "C" void kernel_launch(void* const* d_in, const int* in_sizes, int n_in,
                              void* d_out, int out_size, void* d_ws, size_t ws_size,
                              hipStream_t stream)
{
    const float* x     = (const float*)d_in[0];
    const int*   ei    = (const int*)  d_in[1];
    const int*   batch = (const int*)  d_in[2];
    const float* w_in  = (const float*)d_in[3];
    const float* b_in  = (const float*)d_in[4];
    const float* W[3]    = {(const float*)d_in[5],  (const float*)d_in[9],  (const float*)d_in[13]};
    const float* ASRC[3] = {(const float*)d_in[6],  (const float*)d_in[10], (const float*)d_in[14]};
    const float* ADST[3] = {(const float*)d_in[7],  (const float*)d_in[11], (const float*)d_in[15]};
    const float* BI[3]   = {(const float*)d_in[8],  (const float*)d_in[12], (const float*)d_in[16]};
    const float* w_out = (const float*)d_in[17];
    const float* b_out = (const float*)d_in[18];

    const int N  = in_sizes[0] / 64;   // 50000
    const int E  = in_sizes[1] / 2;    // 800000
    const int Ep = E + N;              // with self-loops

    // workspace carve-out (256B aligned slices)
    char* ws = (char*)d_ws;
    size_t off = 0;
    auto carve = [&](size_t bytes) -> void* {
        void* p = ws + off;
        off += (bytes + 255) & ~(size_t)255;
        return p;
    };
    float*    featA  = (float*)   carve((size_t)N * 128 * 4);
    float*    hW     = (float*)   carve((size_t)N * 128 * 4);
    float*    acc    = (float*)   carve((size_t)N * 128 * 4);
    float*    als    = (float*)   carve((size_t)N * 4 * 4);
    float*    ald    = (float*)   carve((size_t)N * 4 * 4);
    unsigned* menc   = (unsigned*)carve((size_t)N * 4 * 4);
    float*    ssum   = (float*)   carve((size_t)N * 4 * 4);
    float*    ebuf   = (float*)   carve((size_t)Ep * 4 * 4);
    float*    psum   = (float*)   carve(64 * 128 * 4);
    float*    cnt    = (float*)   carve(64 * 4);
    float*    pooled = (float*)   carve(64 * 128 * 4);
    (void)ws_size; (void)n_in;

    const dim3 blk(256);

    // input projection: featA = relu(x @ w_in + b_in)   [N,64]@[64,128]
    gemm_wmma_f32<<<dim3(cdiv(N / 16, 8), 128 / 16), blk, 0, stream>>>(
        x, w_in, b_in, featA, N, 64, 128, 1);

    for (int l = 0; l < 3; ++l) {
        const int heads   = (l < 2) ? 4 : 1;
        const int hshift  = (l < 2) ? 2 : 0;
        const int oc      = 128 / heads;
        const int ocshift = (l < 2) ? 5 : 7;

        // hW = featA @ W[l]  (bias deferred to post-aggregation)
        gemm_wmma_f32<<<dim3(cdiv(N / 16, 8), 128 / 16), blk, 0, stream>>>(
            featA, W[l], nullptr, hW, N, 128, 128, 0);

        attn_coef<<<cdiv((long long)N * heads, 256), blk, 0, stream>>>(
            hW, ASRC[l], ADST[l], als, ald, N, heads, oc, hshift);

        zero_u32<<<cdiv((long long)N * heads, 256), blk, 0, stream>>>(menc, N * heads);
        zero_u32<<<cdiv((long long)N * heads, 256), blk, 0, stream>>>((unsigned*)ssum, N * heads);
        zero_u32<<<cdiv((long long)N * 128, 256), blk, 0, stream>>>((unsigned*)acc, N * 128);

        edge_max<<<cdiv((long long)Ep * heads, 256), blk, 0, stream>>>(
            ei, E, Ep, als, ald, menc, ebuf, heads, hshift);
        edge_exp_sum<<<cdiv((long long)Ep * heads, 256), blk, 0, stream>>>(
            ei, E, Ep, menc, ebuf, ssum, heads, hshift);
        edge_aggregate<<<cdiv((long long)Ep * 32, 256), blk, 0, stream>>>(
            ei, E, Ep, hW, ebuf, ssum, acc, heads, ocshift);

        bias_act<<<cdiv((long long)N * 128, 256), blk, 0, stream>>>(
            acc, BI[l], featA, N * 128, (l < 2) ? 1 : 0);
    }

    // global mean pool over batch segments
    zero_u32<<<cdiv(64 * 128 + 64, 256), blk, 0, stream>>>((unsigned*)psum, 64 * 128 + 64);
    pool_sum<<<cdiv((long long)N * 128, 256), blk, 0, stream>>>(featA, batch, psum, cnt, N);
    pool_div<<<cdiv(64 * 128, 256), blk, 0, stream>>>(psum, cnt, pooled);

    // output projection: d_out = pooled @ w_out + b_out   [64,128]@[128,64]
    gemm_wmma_f32<<<dim3(1, 64 / 16), blk, 0, stream>>>(
        pooled, w_out, b_out, (float*)d_out, 64, 128, 64, 0);
}